// FlowAlign_14611478741522
// MI455X (gfx1250) — compile-verified
//
#include <hip/hip_runtime.h>

// FlowAlign fused kernel for MI455X (gfx1250, wave32).
// Stage 1: 4x4/stride-4 conv of flows (general [2,2,4,4] weights) -> per-pixel flow.
// Stage 2: bilinear warp of 256 feature channels by that flow.
// Flow tile (16 KiB per block) is staged to LDS via the CDNA5 Tensor Data Mover.
// Output stores are non-temporal so the 128 MiB feature set stays L2-resident
// (features 128 MiB + output 128 MiB > 192 MB L2; gathers re-read features ~4x).

namespace {
constexpr int Bn  = 8;
constexpr int Cn  = 256;
constexpr int Hn  = 128;
constexpr int Wn  = 128;
constexpr int FHn = 512;
constexpr int FWn = 512;
constexpr int INVn = 4;
}

typedef __attribute__((ext_vector_type(4))) unsigned int v4u;
typedef __attribute__((ext_vector_type(8))) int          v8i;
typedef __attribute__((ext_vector_type(4))) int          v4i;

#if __has_builtin(__builtin_amdgcn_tensor_load_to_lds) && __has_builtin(__builtin_amdgcn_s_wait_tensorcnt)
#define USE_TDM 1
#else
#define USE_TDM 0
#endif

__global__ __launch_bounds__(256) void flowalign_fused(
    const float* __restrict__ features,   // [8,256,128,128]
    const float* __restrict__ flows,      // [8,2,512,512]
    const float* __restrict__ conv_w,     // [2,2,4,4]
    float* __restrict__ out)              // [8,256,128,128]
{
  const int y   = blockIdx.x;            // 0..127 output row
  const int b   = blockIdx.y;            // 0..7 batch
  const int tid = threadIdx.x;           // 0..255
  const int x   = tid & (Wn - 1);        // pixel column
  const int cg  = tid >> 7;              // channel group: 0 or 1

  // Flow tile: [2 channels][4 rows][512 cols] = 16 KiB, plus conv weights.
  __shared__ __align__(16) float flowLds[2 * INVn * FWn];
  __shared__ float cw[2 * 2 * INVn * INVn];

#if USE_TDM
  // ---- CDNA5 Tensor Data Mover: DMA flows[b, 0:2, 4y:4y+4, 0:512] -> LDS ----
  if (tid < 32) {  // single wave issues the descriptor (TDM ignores EXEC; branch skips other waves)
    const unsigned long long ga =
        (unsigned long long)(uintptr_t)flows +
        4ull * ((unsigned long long)b * 2ull * FHn * FWn +
                (unsigned long long)(y * INVn) * FWn);
    const unsigned int ldsa = (unsigned int)(uintptr_t)&flowLds[0];

    // D# group 0: count=1 | lds_addr | global_addr[56:0] | type=2
    v4u g0;
    g0[0] = 1u;
    g0[1] = ldsa;
    g0[2] = (unsigned int)ga;
    g0[3] = ((unsigned int)(ga >> 32) & 0x01FFFFFFu) | (2u << 30);

    // D# group 1: mask=0, data_size=2 (4B), no pad/iterate/barrier.
    // tensor_dim0=512, tensor_dim1=4 ; tile_dim0=512, tile_dim1=4, tile_dim2=2
    // tensor_dim0_stride=512 (row), tensor_dim1_stride=512*512 (channel plane)
    v8i g1;
    g1[0] = (int)(2u << 16);                                   // data_size=2 -> 4 bytes
    g1[1] = (int)(((unsigned)FWn & 0xFFFFu) << 16);            // tensor_dim0[15:0]
    g1[2] = (int)(((unsigned)FWn >> 16) | ((unsigned)INVn << 16)); // dim0 hi | tensor_dim1 lo
    g1[3] = (int)(((unsigned)FWn & 0xFFFFu) << 16);            // dim1 hi(0) | tile_dim0=512
    g1[4] = (int)((unsigned)INVn | (2u << 16));                // tile_dim1=4 | tile_dim2=2
    g1[5] = (int)FWn;                                          // dim0_stride[31:0]=512
    g1[6] = (int)((((unsigned)(FHn * FWn)) & 0xFFFFu) << 16);  // dim0_stride hi(0) | dim1_stride lo16
    g1[7] = (int)(((unsigned)(FHn * FWn)) >> 16);              // dim1_stride[47:16]

    v4i g2; g2[0] = 2; g2[1] = 0; g2[2] = 0; g2[3] = 0;        // tensor_dim2=2
    v4i g3; g3[0] = 0; g3[1] = 0; g3[2] = 0; g3[3] = 0;

#if defined(__clang_major__) && __clang_major__ >= 23
    v8i gz; gz[0]=0; gz[1]=0; gz[2]=0; gz[3]=0; gz[4]=0; gz[5]=0; gz[6]=0; gz[7]=0;
    __builtin_amdgcn_tensor_load_to_lds(g0, g1, g2, g3, gz, 0);
#else
    __builtin_amdgcn_tensor_load_to_lds(g0, g1, g2, g3, 0);
#endif
    __builtin_amdgcn_s_wait_tensorcnt(0);
  }
#else
  // Fallback: cooperative vectorized staging (2 channels x 4 contiguous rows each)
  for (int idx = tid; idx < 2 * INVn * FWn / 4; idx += 256) {
    const int perCh = INVn * FWn / 4;                 // 512 float4 per channel chunk
    const int c   = idx / perCh;
    const int off = idx - c * perCh;
    const float4* src = (const float4*)(flows +
        ((size_t)b * 2 + c) * (size_t)(FHn * FWn) + (size_t)(y * INVn) * FWn);
    ((float4*)flowLds)[idx] = src[off];
  }
#endif

  if (tid < 64) cw[tid] = conv_w[tid];
  __syncthreads();

  // ---- Stage 1: general 2-in/2-out 4x4 conv at this pixel ----
  const int xb = x * INVn;
  float fx = 0.f, fy = 0.f;
#pragma unroll
  for (int i = 0; i < 2; ++i) {
#pragma unroll
    for (int r = 0; r < INVn; ++r) {
#pragma unroll
      for (int k = 0; k < INVn; ++k) {
        const float v = flowLds[(i * INVn + r) * FWn + xb + k];
        fx = fmaf(cw[((0 * 2 + i) * INVn + r) * INVn + k], v, fx);
        fy = fmaf(cw[((1 * 2 + i) * INVn + r) * INVn + k], v, fy);
      }
    }
  }

  // ---- Stage 2: bilinear warp setup (once per pixel, reused for 128 channels) ----
  const float gx  = (float)x + fx;
  const float gy  = (float)y + fy;
  const float x0f = floorf(gx), y0f = floorf(gy);
  const float wx  = gx - x0f,  wy  = gy - y0f;
  const int x0 = (int)x0f, y0 = (int)y0f;
  const int x1 = x0 + 1,   y1 = y0 + 1;

  const bool vx0 = (x0 >= 0) & (x0 < Wn);
  const bool vx1 = (x1 >= 0) & (x1 < Wn);
  const bool vy0 = (y0 >= 0) & (y0 < Hn);
  const bool vy1 = (y1 >= 0) & (y1 < Hn);

  const int cx0 = min(max(x0, 0), Wn - 1);
  const int cx1 = min(max(x1, 0), Wn - 1);
  const int cy0 = min(max(y0, 0), Hn - 1);
  const int cy1 = min(max(y1, 0), Hn - 1);

  const int i00 = cy0 * Wn + cx0;
  const int i10 = cy0 * Wn + cx1;
  const int i01 = cy1 * Wn + cx0;
  const int i11 = cy1 * Wn + cx1;

  const float w00 = (1.f - wx) * (1.f - wy) * ((vx0 && vy0) ? 1.f : 0.f);
  const float w10 = wx * (1.f - wy)         * ((vx1 && vy0) ? 1.f : 0.f);
  const float w01 = (1.f - wx) * wy         * ((vx0 && vy1) ? 1.f : 0.f);
  const float w11 = wx * wy                 * ((vx1 && vy1) ? 1.f : 0.f);

  // ---- Channel loop: 4 L2-resident gathers + 1 NT store per channel ----
  constexpr int plane = Hn * Wn;
  const float* fbase = features + (size_t)b * Cn * plane;
  float*       obase = out      + (size_t)b * Cn * plane + (size_t)y * Wn + x;

  const int c0 = cg * (Cn / 2);
#pragma unroll 4
  for (int c = c0; c < c0 + Cn / 2; ++c) {
    const float* fp = fbase + (size_t)c * plane;
    const float v = w00 * fp[i00] + w10 * fp[i10] + w01 * fp[i01] + w11 * fp[i11];
    __builtin_nontemporal_store(v, obase + (size_t)c * plane);
  }
}

extern "C" void kernel_launch(void* const* d_in, const int* in_sizes, int n_in,
                              void* d_out, int out_size, void* d_ws, size_t ws_size,
                              hipStream_t stream) {
  (void)in_sizes; (void)n_in; (void)out_size; (void)d_ws; (void)ws_size;
  const float* features = (const float*)d_in[0];
  const float* flows    = (const float*)d_in[1];
  const float* conv_w   = (const float*)d_in[2];
  float* outp           = (float*)d_out;

  dim3 grid(Hn, Bn);   // one block per (output row, batch)
  flowalign_fused<<<grid, 256, 0, stream>>>(features, flows, conv_w, outp);
}